// SelectiveSSMLayer_1090921693696
// MI455X (gfx1250) — compile-verified
//
#include <hip/hip_runtime.h>
#include <math.h>
#include <stdint.h>

// ---------------------------------------------------------------------------
// Problem constants (from reference): B=32, L=2048, D=256, N=16
// ---------------------------------------------------------------------------
constexpr int Bsz   = 32;
constexpr int SeqL  = 2048;
constexpr int Dm    = 256;
constexpr int Ns    = 16;
constexpr int Mrows = Bsz * SeqL; // 65536 token rows
constexpr int CH    = 128;        // scan timesteps per TDM-staged chunk

typedef __attribute__((ext_vector_type(2))) float v2f;
typedef __attribute__((ext_vector_type(8))) float v8f;
typedef __attribute__((ext_vector_type(4))) unsigned int v4u;
typedef __attribute__((ext_vector_type(8))) int v8i;
typedef __attribute__((ext_vector_type(4))) int v4i;

// ---------------------------------------------------------------------------
// Register-blocked WMMA f32 16x16x4 micro-kernel: one wave computes NT
// adjacent 16x16 tiles (16 rows x NT*16 cols). A fragment loaded once per
// K-step and reused across NT wmma's -> 1 + NT loads per NT wmma.
//   arow    = &A[(m0 + lane%16)*Dm + 2*(lane/16)]
//   brow[j] = &W[(n0 + j*16 + lane%16)*Dm + 2*(lane/16)]  (B^T row = W row)
// ---------------------------------------------------------------------------
template <int NT>
__device__ inline void wmma_block(const float* __restrict__ arow,
                                  const float* const* __restrict__ brow,
                                  v8f* __restrict__ acc) {
#pragma unroll 4
  for (int k = 0; k < Dm; k += 4) {
    v2f a = *(const v2f*)(arow + k);
#pragma unroll
    for (int j = 0; j < NT; ++j) {
      v2f b = *(const v2f*)(brow[j] + k);
      acc[j] = __builtin_amdgcn_wmma_f32_16x16x4_f32(
          false, a, false, b, (short)0, acc[j], false, false);
    }
  }
}

// ---------------------------------------------------------------------------
// TDM: 1-D contiguous tile (nelem f32) global -> LDS. Descriptor per
// CDNA5 ISA: group0 = {count|flags, lds_addr, gaddr_lo,
// gaddr_hi|type=2}; group1 packs data_size=4B, tensor/tile dims, strides.
// Issue from ONE wave only (EXEC is ignored by TDM). Tracked by TENSORcnt.
// This toolchain's builtin is the 6-arg form (extra int32x8 group, cpol).
// ---------------------------------------------------------------------------
__device__ inline void tdm_load_1d(const float* gptr, uint32_t lds_off,
                                   uint32_t nelem) {
  uint64_t ga = (uint64_t)(uintptr_t)gptr;
  v4u g0;
  g0[0] = 1u;                                    // count=1, user mode
  g0[1] = lds_off;                               // LDS byte address
  g0[2] = (uint32_t)ga;                          // global_addr[31:0]
  g0[3] = ((uint32_t)(ga >> 32) & 0x01FFFFFFu)   // global_addr[56:32]
        | 0x80000000u;                           // type=2 (bits 127:126)
  v8i g1;
  g1[0] = 0x20000;                               // data_size=2 (4 bytes)
  g1[1] = (int)((nelem & 0xFFFFu) << 16);        // tensor_dim0[15:0]
  g1[2] = (int)(((nelem >> 16) & 0xFFFFu)        // tensor_dim0[31:16]
        | (1u << 16));                           // tensor_dim1 = 1
  g1[3] = (int)((nelem & 0xFFFFu) << 16);        // tile_dim0 (16-bit)
  g1[4] = 0;                                     // tile_dim1/2 unused
  g1[5] = (int)nelem;                            // tensor_dim0_stride lo
  g1[6] = (int)((nelem & 0xFFFFu) << 16);        // stride0 hi=0 | stride1 lo
  g1[7] = (int)((nelem >> 16) & 0xFFFFu);        // stride1 hi
  v4i g2 = {0, 0, 0, 0};
  v4i g3 = {0, 0, 0, 0};
  v8i g4 = {0, 0, 0, 0, 0, 0, 0, 0};
  __builtin_amdgcn_tensor_load_to_lds(g0, g1, g2, g3, g4, 0);
}

// ---------------------------------------------------------------------------
// Kernel 1: xz = x @ W_in^T + b_in ; split into x_in (cols 0..255) and z.
// One wave per 16x64 output block (NT=4). 512 cols -> 8 col groups.
// ---------------------------------------------------------------------------
__global__ void in_proj_kernel(const float* __restrict__ x,
                               const float* __restrict__ W_in,
                               const float* __restrict__ b_in,
                               float* __restrict__ xin,
                               float* __restrict__ zbuf) {
  int wave = (blockIdx.x * blockDim.x + threadIdx.x) >> 5;
  int lane = threadIdx.x & 31;
  int lm   = lane & 15;
  int half = lane >> 4;
  int cg = wave & 7;        // 8 col groups of 64
  int rt = wave >> 3;       // 4096 row tiles
  int m0 = rt * 16;
  int n0 = cg * 64;

  const float* arow = x + (size_t)(m0 + lm) * Dm + 2 * half;
  const float* brow[4];
#pragma unroll
  for (int j = 0; j < 4; ++j)
    brow[j] = W_in + (size_t)(n0 + j * 16 + lm) * Dm + 2 * half;

  v8f acc[4] = {};
  wmma_block<4>(arow, brow, acc);

#pragma unroll
  for (int j = 0; j < 4; ++j) {
    int col = n0 + j * 16 + lm;
    float bias = b_in[col];
    float* dst;
    int c;
    if (col < Dm) { dst = xin;  c = col; }
    else          { dst = zbuf; c = col - Dm; }
#pragma unroll
    for (int v = 0; v < 8; ++v) {
      int row = m0 + v + 8 * half;
      dst[(size_t)row * Dm + c] = acc[j][v] + bias;
    }
  }
}

// ---------------------------------------------------------------------------
// Kernel 2: depthwise conv1d (k=3, pad=1) + bias + SiLU.  One thread/elem.
// ---------------------------------------------------------------------------
__global__ void conv_silu_kernel(const float* __restrict__ xin,
                                 const float* __restrict__ cw,
                                 const float* __restrict__ cb,
                                 float* __restrict__ xconv) {
  size_t idx = (size_t)blockIdx.x * blockDim.x + threadIdx.x;
  int d = (int)(idx & (Dm - 1));
  size_t m = idx >> 8;
  int l = (int)(m & (SeqL - 1));
  float c0 = cw[d * 3 + 0];
  float c1 = cw[d * 3 + 1];
  float c2 = cw[d * 3 + 2];
  float acc = cb[d] + c1 * xin[m * Dm + d];
  if (l > 0)        acc += c0 * xin[(m - 1) * Dm + d];
  if (l < SeqL - 1) acc += c2 * xin[(m + 1) * Dm + d];
  float sg = 1.0f / (1.0f + __expf(-acc));
  xconv[m * Dm + d] = acc * sg;
}

// ---------------------------------------------------------------------------
// Kernel 3: fused projections off x_conv. Col groups per row tile:
//   g=0..3  -> delta tiles (NT=4, cols g*64..g*64+63), softplus + b_dt
//   g=4     -> B_sel and C_sel (NT=2)
// ---------------------------------------------------------------------------
__global__ void proj_dtbc_kernel(const float* __restrict__ xconv,
                                 const float* __restrict__ W_dt,
                                 const float* __restrict__ b_dt,
                                 const float* __restrict__ W_B,
                                 const float* __restrict__ W_C,
                                 float* __restrict__ delta,
                                 float* __restrict__ bsel,
                                 float* __restrict__ csel) {
  int wave = (blockIdx.x * blockDim.x + threadIdx.x) >> 5;
  int lane = threadIdx.x & 31;
  int lm   = lane & 15;
  int half = lane >> 4;
  int g  = wave % 5;
  int rt = wave / 5;
  int m0 = rt * 16;

  const float* arow = xconv + (size_t)(m0 + lm) * Dm + 2 * half;

  if (g < 4) {
    int n0 = g * 64;
    const float* brow[4];
#pragma unroll
    for (int j = 0; j < 4; ++j)
      brow[j] = W_dt + (size_t)(n0 + j * 16 + lm) * Dm + 2 * half;
    v8f acc[4] = {};
    wmma_block<4>(arow, brow, acc);
#pragma unroll
    for (int j = 0; j < 4; ++j) {
      int col = n0 + j * 16 + lm;
      float bias = b_dt[col];
#pragma unroll
      for (int v = 0; v < 8; ++v) {
        int row = m0 + v + 8 * half;
        float t = acc[j][v] + bias;
        float sp = (t > 20.0f) ? t : log1pf(__expf(t));  // stable softplus
        delta[(size_t)row * Dm + col] = sp;
      }
    }
  } else {
    const float* brow[2];
    brow[0] = W_B + (size_t)lm * Dm + 2 * half;
    brow[1] = W_C + (size_t)lm * Dm + 2 * half;
    v8f acc[2] = {};
    wmma_block<2>(arow, brow, acc);
#pragma unroll
    for (int j = 0; j < 2; ++j) {
      float* dst = (j == 0) ? bsel : csel;
#pragma unroll
      for (int v = 0; v < 8; ++v) {
        int row = m0 + v + 8 * half;
        dst[(size_t)row * Ns + lm] = acc[j][v];
      }
    }
  }
}

// ---------------------------------------------------------------------------
// Kernel 4: selective scan + skip(D) + SiLU(z) gate.
// One block per batch b (256 lanes = 256 channels d). B_sel/C_sel rows are
// wave-uniform, so they are staged by the Tensor Data Mover into LDS
// (double-buffered, CH=128 steps = 8KB per array), issued by wave 0 only,
// synchronized with s_wait_tensorcnt + workgroup barrier. Per-step reads
// then come from LDS (broadcast), dt/xc/z stay coalesced global loads.
// ---------------------------------------------------------------------------
__global__ void scan_kernel(const float* __restrict__ delta,
                            const float* __restrict__ xconv,
                            const float* __restrict__ zbuf,
                            const float* __restrict__ bsel,
                            const float* __restrict__ csel,
                            const float* __restrict__ A_log,
                            const float* __restrict__ D_param,
                            float* __restrict__ yg) {
  __shared__ alignas(16) float sB[2][CH * Ns];
  __shared__ alignas(16) float sC[2][CH * Ns];

  int d = threadIdx.x;      // 0..255 channel
  int b = blockIdx.x;       // 0..31 batch
  bool issuer = (threadIdx.x < 32);  // wave 0 (wave-uniform branch)

  float A[Ns], h[Ns];
#pragma unroll
  for (int n = 0; n < Ns; ++n) {
    A[n] = -__expf(A_log[d * Ns + n]);
    h[n] = 0.0f;
  }
  float Dp = D_param[d];

  size_t base = (size_t)b * SeqL;
  constexpr int NCH = SeqL / CH;  // 16 chunks

  if (issuer) {
    tdm_load_1d(bsel + base * Ns, (uint32_t)(uintptr_t)&sB[0][0], CH * Ns);
    tdm_load_1d(csel + base * Ns, (uint32_t)(uintptr_t)&sC[0][0], CH * Ns);
  }

  for (int c = 0; c < NCH; ++c) {
    int buf = c & 1;
    if (issuer) {
      if (c + 1 < NCH) {
        size_t nb = (base + (size_t)(c + 1) * CH) * Ns;
        tdm_load_1d(bsel + nb, (uint32_t)(uintptr_t)&sB[buf ^ 1][0], CH * Ns);
        tdm_load_1d(csel + nb, (uint32_t)(uintptr_t)&sC[buf ^ 1][0], CH * Ns);
        __builtin_amdgcn_s_wait_tensorcnt(2);  // chunk c's 2 ops done
      } else {
        __builtin_amdgcn_s_wait_tensorcnt(0);
      }
    }
    __syncthreads();  // publish staged chunk c to all waves

    for (int t = 0; t < CH; ++t) {
      size_t r = base + (size_t)c * CH + t;
      float dt = delta[r * Dm + d];
      float xc = xconv[r * Dm + d];
      float zz = zbuf[r * Dm + d];
      if (t + 4 < CH) {  // stream-prefetch ahead (1 KB row stride)
        __builtin_prefetch(delta + (r + 4) * Dm + d, 0, 0);
        __builtin_prefetch(xconv + (r + 4) * Dm + d, 0, 0);
        __builtin_prefetch(zbuf + (r + 4) * Dm + d, 0, 0);
      }

      float Bv[Ns], Cv[Ns];
      const float4* pb = (const float4*)&sB[buf][t * Ns];
      const float4* pc = (const float4*)&sC[buf][t * Ns];
#pragma unroll
      for (int q = 0; q < 4; ++q) {
        float4 qb = pb[q];
        float4 qc = pc[q];
        Bv[4 * q + 0] = qb.x; Bv[4 * q + 1] = qb.y;
        Bv[4 * q + 2] = qb.z; Bv[4 * q + 3] = qb.w;
        Cv[4 * q + 0] = qc.x; Cv[4 * q + 1] = qc.y;
        Cv[4 * q + 2] = qc.z; Cv[4 * q + 3] = qc.w;
      }

      float dx = dt * xc;
      float y = 0.0f;
#pragma unroll
      for (int n = 0; n < Ns; ++n) {
        float ab = __expf(dt * A[n]);                 // exp(dt*A)
        h[n] = __builtin_fmaf(ab, h[n], dx * Bv[n]);  // h = ab*h + dt*xc*B
        y = __builtin_fmaf(h[n], Cv[n], y);           // y += h*C
      }
      y = __builtin_fmaf(xc, Dp, y);                  // + x_conv * D
      float sg = 1.0f / (1.0f + __expf(-zz));
      yg[r * Dm + d] = y * (zz * sg);                 // * SiLU(z)
    }
    __syncthreads();  // all reads of buf done before it is re-filled
  }
}

// ---------------------------------------------------------------------------
// Kernel 5: out = yg @ W_out^T + b_out  (NT=4, writes d_out directly)
// ---------------------------------------------------------------------------
__global__ void out_proj_kernel(const float* __restrict__ yg,
                                const float* __restrict__ W_out,
                                const float* __restrict__ b_out,
                                float* __restrict__ out) {
  int wave = (blockIdx.x * blockDim.x + threadIdx.x) >> 5;
  int lane = threadIdx.x & 31;
  int lm   = lane & 15;
  int half = lane >> 4;
  int cg = wave & 3;        // 4 col groups of 64
  int rt = wave >> 2;
  int m0 = rt * 16;
  int n0 = cg * 64;

  const float* arow = yg + (size_t)(m0 + lm) * Dm + 2 * half;
  const float* brow[4];
#pragma unroll
  for (int j = 0; j < 4; ++j)
    brow[j] = W_out + (size_t)(n0 + j * 16 + lm) * Dm + 2 * half;

  v8f acc[4] = {};
  wmma_block<4>(arow, brow, acc);

#pragma unroll
  for (int j = 0; j < 4; ++j) {
    int col = n0 + j * 16 + lm;
    float bias = b_out[col];
#pragma unroll
    for (int v = 0; v < 8; ++v) {
      int row = m0 + v + 8 * half;
      out[(size_t)row * Dm + col] = acc[j][v] + bias;
    }
  }
}

// ---------------------------------------------------------------------------
// Host launcher. Input order: x, A_log, D_param, W_B, W_C, W_dt, b_dt,
// W_in, b_in, W_out, b_out, conv_w, conv_b. Output: [B, L, D] f32.
// ---------------------------------------------------------------------------
extern "C" void kernel_launch(void* const* d_in, const int* in_sizes, int n_in,
                              void* d_out, int out_size, void* d_ws, size_t ws_size,
                              hipStream_t stream) {
  (void)in_sizes; (void)n_in; (void)out_size; (void)ws_size;

  const float* x      = (const float*)d_in[0];
  const float* A_log  = (const float*)d_in[1];
  const float* Dpar   = (const float*)d_in[2];
  const float* W_B    = (const float*)d_in[3];
  const float* W_C    = (const float*)d_in[4];
  const float* W_dt   = (const float*)d_in[5];
  const float* b_dt   = (const float*)d_in[6];
  const float* W_in   = (const float*)d_in[7];
  const float* b_in   = (const float*)d_in[8];
  const float* W_out  = (const float*)d_in[9];
  const float* b_out  = (const float*)d_in[10];
  const float* conv_w = (const float*)d_in[11];
  const float* conv_b = (const float*)d_in[12];
  float* out = (float*)d_out;

  // Workspace layout (floats): 4 x [M,D] + 2 x [M,N]; yg reuses xin.
  float* ws    = (float*)d_ws;
  const size_t MD = (size_t)Mrows * Dm;
  float* xin   = ws;
  float* zbuf  = xin   + MD;
  float* xconv = zbuf  + MD;
  float* delta = xconv + MD;
  float* bsel  = delta + MD;
  float* csel  = bsel  + (size_t)Mrows * Ns;
  float* yg    = xin;   // xin dead after conv

  // 1) in_proj: (M/16) * (512/64) = 32768 waves
  in_proj_kernel<<<4096, 256, 0, stream>>>(x, W_in, b_in, xin, zbuf);
  // 2) depthwise conv + SiLU
  conv_silu_kernel<<<(Mrows * Dm) / 256, 256, 0, stream>>>(xin, conv_w, conv_b, xconv);
  // 3) delta/B/C projections: (M/16) * 5 = 20480 waves
  proj_dtbc_kernel<<<2560, 256, 0, stream>>>(xconv, W_dt, b_dt, W_B, W_C,
                                             delta, bsel, csel);
  // 4) selective scan + gate: one block per batch
  scan_kernel<<<Bsz, Dm, 0, stream>>>(delta, xconv, zbuf, bsel, csel,
                                      A_log, Dpar, yg);
  // 5) out_proj: (M/16) * (256/64) = 16384 waves
  out_proj_kernel<<<2048, 256, 0, stream>>>(yg, W_out, b_out, out);
}